// RuntimeConv_953482740315
// MI455X (gfx1250) — compile-verified
//
#include <hip/hip_runtime.h>
#include <math.h>

typedef float v2f __attribute__((ext_vector_type(2)));
typedef float v8f __attribute__((ext_vector_type(8)));

#define IN_CH 128
#define HID   64
#define OUTC  8

__device__ __forceinline__ float selu_f(float x) {
    const float scale = 1.0507009873554804934193349852946f;
    const float alpha = 1.6732632423543772848170429916717f;
    return x > 0.0f ? scale * x : scale * alpha * (expf(x) - 1.0f);
}

// ---------------------------------------------------------------------------
// Kernel 0: pack W1/W2 into WMMA B-fragment lane order so each lane of the
// GEMM loop loads one aligned float2 (global_load_b64) per fragment.
// w1p[kstep*128 + j*32 + lane] = { W1[ksub][col], W1[ksub+1][col] }
//   ksub = kstep*4 + (lane>>4)*2, col = j*16 + (lane&15)
// w2p[kstep*32 + lane] = same for W2 zero-padded to 16 cols.
// ---------------------------------------------------------------------------
__global__ __launch_bounds__(256)
void pack_kernel(const float* __restrict__ W1, const float* __restrict__ W2,
                 float2* __restrict__ w1p, float2* __restrict__ w2p) {
    int t = blockIdx.x * 256 + threadIdx.x;
    if (t < 32 * 4 * 32) {                       // 4096 W1 fragments
        int kstep = t >> 7;
        int j     = (t >> 5) & 3;
        int lane  = t & 31;
        int ksub  = kstep * 4 + (lane >> 4) * 2;
        int col   = j * 16 + (lane & 15);
        w1p[t] = make_float2(W1[ksub * HID + col], W1[(ksub + 1) * HID + col]);
    } else if (t < 32 * 4 * 32 + 16 * 32) {      // 512 W2 fragments
        int u = t - 32 * 4 * 32;
        int kstep = u >> 5;
        int lane  = u & 31;
        int ksub  = kstep * 4 + (lane >> 4) * 2;
        int col   = lane & 15;
        float2 v = make_float2(0.0f, 0.0f);
        if (col < OUTC)
            v = make_float2(W2[ksub * OUTC + col], W2[(ksub + 1) * OUTC + col]);
        w2p[u] = v;
    }
}

// ---------------------------------------------------------------------------
// Kernel 1: init agg (= second half of d_out) to -inf bits, flags to 0
// ---------------------------------------------------------------------------
__global__ __launch_bounds__(256)
void init_kernel(int* agg_bits, int* flags, int n8, int n) {
    int i = blockIdx.x * 256 + threadIdx.x;
    if (i < n8) agg_bits[i] = 0xFF800000;        // -inf
    if (i < n)  flags[i] = 0;
}

// ---------------------------------------------------------------------------
// Layer-1 segment helper: straight-line K loop over one feature array.
// No branches in the hot path; all loads are base + immediate offset.
// ---------------------------------------------------------------------------
template<int KSTART, int KLEN, int STRIDE>
__device__ __forceinline__ void l1_seg(const float* __restrict__ base,
                                       int rowA, int half, int lane,
                                       const float2* __restrict__ w1p,
                                       v8f acc[4]) {
    const float* pa = base + (size_t)rowA * STRIDE + half * 2;
    #pragma unroll
    for (int kk = 0; kk < KLEN; kk += 4) {
        v2f a = *(const v2f*)(pa + kk);          // aligned: even offset
        const float2* pb = w1p + ((KSTART + kk) >> 2) * 128 + lane;
        #pragma unroll
        for (int j = 0; j < 4; ++j) {
            float2 bf = pb[j * 32];
            v2f b; b.x = bf.x; b.y = bf.y;
            acc[j] = __builtin_amdgcn_wmma_f32_16x16x4_f32(
                false, a, false, b, (short)0, acc[j], false, false);
        }
    }
}

// ---------------------------------------------------------------------------
// Kernel 2: fused 2-layer SELU MLP via V_WMMA_F32_16X16X4_F32.
// One wave = one 16-row node tile, 8 waves / block, no inter-wave sync.
// ---------------------------------------------------------------------------
__global__ __launch_bounds__(256)
void mlp_kernel(const float* __restrict__ ctx,   // [n,76]
                const float* __restrict__ smx,   // [n,40]
                const float* __restrict__ sev,   // [n,8]
                const float* __restrict__ ovh,   // [n,4]
                const unsigned char* __restrict__ realn, // [n] bool
                const float* __restrict__ cumsum,        // [n,8]
                const float2* __restrict__ w1p,  // packed W1 fragments
                const float2* __restrict__ w2p,  // packed W2 fragments (padded)
                const float* __restrict__ b1,    // [64]
                const float* __restrict__ b2,    // [8]
                float* __restrict__ pred_out,    // [n,8]
                float* __restrict__ sumtab,      // [n,8]
                int n)
{
    __shared__ float sH[8][16 * HID];            // 32 KB: per-wave h staging

    const int lane = threadIdx.x & 31;
    const int wave = threadIdx.x >> 5;
    const int rowBase = (blockIdx.x * 8 + wave) * 16;
    if (rowBase >= n) return;                    // wave-uniform; EXEC stays full

    const int half = lane >> 4;                  // 0: lanes 0-15, 1: lanes 16-31
    const int l16  = lane & 15;

    int rowA = rowBase + l16;                    // clamp tail rows (loads only;
    if (rowA >= n) rowA = n - 1;                 // keeps EXEC all-ones for WMMA)

    // ---------------- Layer 1: [16x128] x [128x64] ----------------
    v8f acc[4];
    #pragma unroll
    for (int j = 0; j < 4; ++j) acc[j] = (v8f){0,0,0,0,0,0,0,0};

    l1_seg<  0, 76, 76>(ctx, rowA, half, lane, w1p, acc);
    l1_seg< 76, 40, 40>(smx, rowA, half, lane, w1p, acc);
    l1_seg<116,  8,  8>(sev, rowA, half, lane, w1p, acc);
    l1_seg<124,  4,  4>(ovh, rowA, half, lane, w1p, acc);

    // selu(acc + b1) -> LDS row-major [16][64] (A-layout for layer 2)
    float* hrow = &sH[wave][0];
    #pragma unroll
    for (int j = 0; j < 4; ++j) {
        #pragma unroll
        for (int r = 0; r < 8; ++r) {
            int m    = r + half * 8;             // D layout: VGPR r -> M=r / r+8
            int ncol = j * 16 + l16;
            hrow[m * HID + ncol] = selu_f(acc[j][r] + b1[ncol]);
        }
    }

    // ---------------- Layer 2: [16x64] x [64x16(pad)] ----------------
    v8f acc2 = (v8f){0,0,0,0,0,0,0,0};
    const float*  ph  = hrow + l16 * HID + half * 2;
    const float2* pb2 = w2p + lane;
    #pragma unroll
    for (int ks = 0; ks < 16; ++ks) {
        v2f a = *(const v2f*)(ph + ks * 4);      // ds_load_b64, aligned
        float2 bf = pb2[ks * 32];
        v2f b; b.x = bf.x; b.y = bf.y;
        acc2 = __builtin_amdgcn_wmma_f32_16x16x4_f32(
            false, a, false, b, (short)0, acc2, false, false);
    }

    // Epilogue: selu + mask, store pred and pred+cumsum table
    #pragma unroll
    for (int r = 0; r < 8; ++r) {
        int m    = r + half * 8;
        int row  = rowBase + m;
        int ncol = l16;
        if (row < n && ncol < OUTC) {
            float v = selu_f(acc2[r] + b2[ncol]);
            v = realn[row] ? v : 0.0f;
            size_t idx = (size_t)row * OUTC + ncol;
            pred_out[idx] = v;
            sumtab[idx]   = v + cumsum[idx];
        }
    }
}

// ---------------------------------------------------------------------------
// Kernel 3: per-edge gather + scatter-max. Sign-aware int atomics lower to
// native global_atomic_max_i32 / global_atomic_min_u32; the 16 MB agg region
// and 16 MB message table are L2-resident (192 MB L2).
// ---------------------------------------------------------------------------
__device__ __forceinline__ void atomicMaxF(float* addr, float v) {
    if (v >= 0.0f) atomicMax((int*)addr, __float_as_int(v));
    else           atomicMin((unsigned int*)addr, __float_as_uint(v));
}

__global__ __launch_bounds__(256)
void edge_kernel(const int* __restrict__ ei,     // [2,E]: src row then dst row
                 const float* __restrict__ sumtab,
                 float* __restrict__ agg,
                 int* __restrict__ flags,
                 int E)
{
    int i = blockIdx.x * 256 + threadIdx.x;
    if (i >= E) return;
    int s = ei[i];
    int d = ei[(size_t)E + i];
    const float4* t = (const float4*)(sumtab + (size_t)s * OUTC);
    float4 m0 = t[0];
    float4 m1 = t[1];
    float* base = agg + (size_t)d * OUTC;
    atomicMaxF(base + 0, m0.x);
    atomicMaxF(base + 1, m0.y);
    atomicMaxF(base + 2, m0.z);
    atomicMaxF(base + 3, m0.w);
    atomicMaxF(base + 4, m1.x);
    atomicMaxF(base + 5, m1.y);
    atomicMaxF(base + 6, m1.z);
    atomicMaxF(base + 7, m1.w);
    flags[d] = 1;                                // benign race: all write 1
}

// ---------------------------------------------------------------------------
// Kernel 4: zero out segments with no incoming edges (in place in d_out)
// ---------------------------------------------------------------------------
__global__ __launch_bounds__(256)
void fin_kernel(float* __restrict__ agg, const int* __restrict__ flags, int n8) {
    int i = blockIdx.x * 256 + threadIdx.x;
    if (i >= n8) return;
    float v = agg[i];
    agg[i] = flags[i >> 3] ? v : 0.0f;
}

// ---------------------------------------------------------------------------
extern "C" void kernel_launch(void* const* d_in, const int* in_sizes, int n_in,
                              void* d_out, int out_size, void* d_ws, size_t ws_size,
                              hipStream_t stream) {
    const int*   ei     = (const int*)d_in[0];            // edge_index [2,E]
    const float* sev    = (const float*)d_in[1];          // [n,8]
    const float* ctx    = (const float*)d_in[2];          // [n,76]
    const float* smx    = (const float*)d_in[3];          // [n,40]
    const unsigned char* realn = (const unsigned char*)d_in[4]; // [n] bool
    const float* ovh    = (const float*)d_in[5];          // [n,4]
    const float* cumsum = (const float*)d_in[6];          // [n,8]
    const float* W1     = (const float*)d_in[7];
    const float* b1     = (const float*)d_in[8];
    const float* W2     = (const float*)d_in[9];
    const float* b2     = (const float*)d_in[10];

    const int n  = in_sizes[6] / OUTC;
    const int E  = in_sizes[0] / 2;
    const int n8 = n * OUTC;

    float* pred = (float*)d_out;                 // [n,8]
    float* agg  = pred + (size_t)n8;             // [n,8] (second output, in place)

    // Workspace layout: sumtab [n8 f32] | flags [n i32] | w1p [4096 f2] | w2p [512 f2]
    float*  sumtab = (float*)d_ws;
    int*    flags  = (int*)((char*)d_ws + (size_t)n8 * sizeof(float));
    float2* w1p    = (float2*)((char*)flags + (size_t)n * sizeof(int));
    float2* w2p    = w1p + 32 * 4 * 32;

    pack_kernel<<<(32 * 4 * 32 + 16 * 32 + 255) / 256, 256, 0, stream>>>(W1, W2, w1p, w2p);
    init_kernel<<<(n8 + 255) / 256, 256, 0, stream>>>((int*)agg, flags, n8, n);

    const int tiles  = (n + 15) / 16;
    const int blocks = (tiles + 7) / 8;
    mlp_kernel<<<blocks, 256, 0, stream>>>(ctx, smx, sev, ovh, realn, cumsum,
                                           w1p, w2p, b1, b2, pred, sumtab, n);

    edge_kernel<<<(E + 255) / 256, 256, 0, stream>>>(ei, sumtab, agg, flags, E);

    fin_kernel<<<(n8 + 255) / 256, 256, 0, stream>>>(agg, flags, n8);
}